// NodeEdgeAggregatorV2_58944131170467
// MI455X (gfx1250) — compile-verified
//
#include <hip/hip_runtime.h>
#include <math.h>

// ---------------- problem constants (match reference) ----------------
static constexpr int   NN   = 50000;     // nodes
static constexpr int   EE   = 800000;    // edges
static constexpr int   LGE  = 1600000;   // line-graph edges
static constexpr int   FND  = 256;       // node feature dim
static constexpr int   TT   = 64;        // edge feature dim
static constexpr int   HID  = 128;
static constexpr int   OUTD = 64;
static constexpr float SLOPE = 0.2f;

typedef float v2f __attribute__((ext_vector_type(2)));
typedef float v8f __attribute__((ext_vector_type(8)));

// =====================================================================
// Fused dual fp32 WMMA GEMM:  C[M,Nc] = act( A1@W1  (+ A2@W2) )
//
// TILESN and DUAL are compile-time so the inner loop is branch-free
// straight-line code: per k-step, issue ALL B-fragment ds_load_b64s,
// then run 8/16 back-to-back v_wmma_f32_16x16x4_f32 on distinct
// accumulators (A read from HBM exactly once per GEMM; W staged in LDS
// transposed [n][k] so B fragments are contiguous b64 loads).
// EXEC discipline: tail waves clamp their tile and still run barriers
// and WMMAs; only the C store is guarded by a wave-uniform branch.
// act: 0=none 1=relu 2=leaky(0.2)
// =====================================================================
static constexpr int KC = 32;

template <int TILESN, bool DUAL>
__global__ __launch_bounds__(256) void k_gemm2_f32_wmma(
    const float* __restrict__ A1, const float* __restrict__ W1,
    const float* __restrict__ A2, const float* __restrict__ W2,
    float* __restrict__ C, int M, int K, int act)
{
    constexpr int Nc = TILESN * 16;
    __shared__ float sW1[Nc * KC];               // transposed: sW[n*KC + k]
    __shared__ float sW2[DUAL ? Nc * KC : 1];

    const int tilesM = M >> 4;
    const int wave = threadIdx.x >> 5;
    const int lane = threadIdx.x & 31;
    int mt = blockIdx.x * 8 + wave;
    const bool valid = (mt < tilesM);
    if (!valid) mt = tilesM - 1;                 // clamp, store is guarded
    const int m0 = mt << 4;
    const int r  = lane & 15;                    // A-row / B,C-col selector
    const int h  = lane >> 4;                    // half-wave selector

    v8f zero = {};
    v8f c[TILESN];
#pragma unroll
    for (int t = 0; t < TILESN; ++t) c[t] = zero;

    const float* __restrict__ a1row = A1 + (size_t)(m0 + r) * K + 2 * h;
    const float* __restrict__ a2row = DUAL ? (A2 + (size_t)(m0 + r) * K + 2 * h) : a1row;

    for (int kb = 0; kb < K; kb += KC) {
        __syncthreads();
        // cooperative staged+transposed copy of W chunks into LDS
#pragma unroll
        for (int it = 0; it < (KC * Nc) / 256; ++it) {
            int idx = it * 256 + threadIdx.x;
            int kk = idx / Nc;                   // Nc constexpr -> shifts
            int nn = idx & (Nc - 1);
            sW1[nn * KC + kk] = W1[(size_t)(kb + kk) * Nc + nn];
            if (DUAL) sW2[nn * KC + kk] = W2[(size_t)(kb + kk) * Nc + nn];
        }
        __syncthreads();

        __builtin_prefetch(a1row + kb + KC, 0, 3);       // next A chunk
        if (DUAL) __builtin_prefetch(a2row + kb + KC, 0, 3);

        for (int k0 = 0; k0 < KC; k0 += 4) {
            v2f a1 = *(const v2f*)(a1row + kb + k0);     // A[m][k0+2h], +1
            v2f a2 = DUAL ? *(const v2f*)(a2row + kb + k0) : a1;
            const int krow = k0 + 2 * h;

            // ---- load phase: issue every B fragment for this k-step ----
            v2f b1[TILESN];
            v2f b2[DUAL ? TILESN : 1];
#pragma unroll
            for (int tn = 0; tn < TILESN; ++tn) {
                b1[tn] = *(const v2f*)(sW1 + (tn * 16 + r) * KC + krow);
                if (DUAL)
                    b2[tn] = *(const v2f*)(sW2 + (tn * 16 + r) * KC + krow);
            }
            // ---- compute phase: back-to-back WMMAs, rotating accums ----
#pragma unroll
            for (int tn = 0; tn < TILESN; ++tn) {
                c[tn] = __builtin_amdgcn_wmma_f32_16x16x4_f32(
                    false, a1, false, b1[tn], (short)0, c[tn], false, false);
                if (DUAL)
                    c[tn] = __builtin_amdgcn_wmma_f32_16x16x4_f32(
                        false, a2, false, b2[tn], (short)0, c[tn], false, false);
            }
        }
    }

    if (valid) {
#pragma unroll
        for (int tn = 0; tn < TILESN; ++tn) {
            const int n0 = tn << 4;
#pragma unroll
            for (int j = 0; j < 8; ++j) {
                float xv = c[tn][j];
                float relu  = fmaxf(xv, 0.f);
                float leaky = xv > 0.f ? xv : SLOPE * xv;
                xv = (act == 1) ? relu : (act == 2) ? leaky : xv;
                C[(size_t)(m0 + 8 * h + j) * Nc + n0 + r] = xv;
            }
        }
    }
}

// ---------------- elementwise / scatter utility kernels ----------------
__global__ void k_fill(float* __restrict__ p, float v, long n)
{
    long i = (long)blockIdx.x * blockDim.x + threadIdx.x;
    if (i < n) p[i] = v;
}

__global__ void k_copy(float* __restrict__ o, const float* __restrict__ s, long n)
{
    long i = (long)blockIdx.x * blockDim.x + threadIdx.x;
    if (i < n) o[i] = s[i];
}

__global__ void k_deg(const int* __restrict__ dst, float* __restrict__ deg, long nE)
{
    long i = (long)blockIdx.x * blockDim.x + threadIdx.x;
    if (i < nE) atomicAdd(&deg[dst[i]], 1.0f);
}

// out[dst[e], :] += h[srcIdx ? srcIdx[e] : e, :]   (float4 per thread)
// F = 4 << qshift
__global__ void k_scatter_add(const float* __restrict__ h,
                              const int* __restrict__ srcIdx,
                              const int* __restrict__ dstIdx,
                              float* __restrict__ out, long nE, int F, int qshift)
{
    long tid = (long)blockIdx.x * blockDim.x + threadIdx.x;
    if (tid >= (nE << qshift)) return;
    long e = tid >> qshift;
    int  q = (int)(tid & ((1 << qshift) - 1));
    int  s = srcIdx ? srcIdx[e] : (int)e;
    int  d = dstIdx[e];
    float4 v = *(const float4*)(h + (size_t)s * F + 4 * q);
    float* o = out + (size_t)d * F + 4 * q;
    atomicAdd(o + 0, v.x); atomicAdd(o + 1, v.y);
    atomicAdd(o + 2, v.z); atomicAdd(o + 3, v.w);
}

// h[row, :] /= max(deg[row], 1)    (F = 1 << fshift)
__global__ void k_div_deg(float* __restrict__ h, const float* __restrict__ deg,
                          long nRows, int fshift)
{
    long i = (long)blockIdx.x * blockDim.x + threadIdx.x;
    if (i < (nRows << fshift)) h[i] /= fmaxf(deg[i >> fshift], 1.0f);
}

// ---------------- attention kernels ----------------
// score[e] = leaky( dot(q[dst[e]], k[src[e]]) / sqrt(HID) )  — wave per edge
__global__ void k_edge_score(const float* __restrict__ q, const float* __restrict__ k,
                             const int* __restrict__ src, const int* __restrict__ dst,
                             float* __restrict__ score, long nE)
{
    long wave = ((long)blockIdx.x * blockDim.x + threadIdx.x) >> 5;
    int  lane = threadIdx.x & 31;
    if (wave >= nE) return;
    int s = src[wave], d = dst[wave];
    float4 qv = *(const float4*)(q + (size_t)d * HID + 4 * lane);
    float4 kv = *(const float4*)(k + (size_t)s * HID + 4 * lane);
    float dot = qv.x * kv.x + qv.y * kv.y + qv.z * kv.z + qv.w * kv.w;
#pragma unroll
    for (int off = 16; off; off >>= 1) dot += __shfl_xor(dot, off, 32);
    if (lane == 0) {
        float sc = dot * 0.08838834764831845f;  // 1/sqrt(128)
        score[wave] = sc > 0.f ? sc : SLOPE * sc;
    }
}

// segment max via int-punned float atomic max (m initialized to -inf)
__global__ void k_seg_max(const float* __restrict__ score, const int* __restrict__ dst,
                          float* __restrict__ m, long nE)
{
    long i = (long)blockIdx.x * blockDim.x + threadIdx.x;
    if (i >= nE) return;
    float v = score[i];
    float* a = &m[dst[i]];
    if (v >= 0.f) atomicMax((int*)a, __float_as_int(v));
    else          atomicMin((unsigned int*)a, (unsigned int)__float_as_int(v));
}

__global__ void k_fix_m(float* __restrict__ m, long n)
{
    long i = (long)blockIdx.x * blockDim.x + threadIdx.x;
    if (i < n) { float v = m[i]; m[i] = isfinite(v) ? v : 0.0f; }
}

__global__ void k_exp_den(const float* __restrict__ score, const int* __restrict__ dst,
                          const float* __restrict__ m, float* __restrict__ ex,
                          float* __restrict__ den, long nE)
{
    long i = (long)blockIdx.x * blockDim.x + threadIdx.x;
    if (i >= nE) return;
    int d = dst[i];
    float e = __expf(score[i] - m[d]);
    ex[i] = e;
    atomicAdd(&den[d], e);
}

// out[dst] += (ex[e]/max(den[dst],1e-9)) * v[src]   (HID=128 -> qshift 5)
__global__ void k_attn_scatter(const float* __restrict__ ex, const float* __restrict__ den,
                               const float* __restrict__ v,
                               const int* __restrict__ src, const int* __restrict__ dst,
                               float* __restrict__ out, long nE)
{
    long tid = (long)blockIdx.x * blockDim.x + threadIdx.x;
    if (tid >= (nE << 5)) return;
    long e = tid >> 5;
    int  q = (int)(tid & 31);
    int  s = src[e], d = dst[e];
    float alpha = ex[e] / fmaxf(den[d], 1e-9f);
    float4 vv = *(const float4*)(v + (size_t)s * HID + 4 * q);
    float* o = out + (size_t)d * HID + 4 * q;
    atomicAdd(o + 0, alpha * vv.x); atomicAdd(o + 1, alpha * vv.y);
    atomicAdd(o + 2, alpha * vv.z); atomicAdd(o + 3, alpha * vv.w);
}

// log_softmax over rows of [N, 64]: wave per row, 2 cols per lane
__global__ void k_logsoftmax(const float* __restrict__ L, float* __restrict__ out, long nRows)
{
    long row = ((long)blockIdx.x * blockDim.x + threadIdx.x) >> 5;
    int lane = threadIdx.x & 31;
    if (row >= nRows) return;
    const float* p = L + (size_t)row * OUTD;
    float a = p[lane], b = p[lane + 32];
    float mx = fmaxf(a, b);
#pragma unroll
    for (int off = 16; off; off >>= 1) mx = fmaxf(mx, __shfl_xor(mx, off, 32));
    float se = __expf(a - mx) + __expf(b - mx);
#pragma unroll
    for (int off = 16; off; off >>= 1) se += __shfl_xor(se, off, 32);
    float ls = __logf(se);
    float* q = out + (size_t)row * OUTD;
    q[lane]      = a - mx - ls;
    q[lane + 32] = b - mx - ls;
}

// =====================================================================
extern "C" void kernel_launch(void* const* d_in, const int* in_sizes, int n_in,
                              void* d_out, int out_size, void* d_ws, size_t ws_size,
                              hipStream_t stream)
{
    (void)in_sizes; (void)n_in; (void)out_size; (void)ws_size;

    const float* x        = (const float*)d_in[0];
    const float* et       = (const float*)d_in[1];
    const int*   Hm       = (const int*)d_in[2];
    const int*   rei      = (const int*)d_in[3];
    const int*   lgei     = (const int*)d_in[4];
    const float* W_tsa1_s = (const float*)d_in[5];
    const float* W_tsa1_n = (const float*)d_in[6];
    const float* W_tsa2_s = (const float*)d_in[7];
    const float* W_tsa2_n = (const float*)d_in[8];
    const float* W_etn    = (const float*)d_in[9];
    const float* W_eg_lin = (const float*)d_in[10];
    const float* W_ea_s   = (const float*)d_in[11];
    const float* W_ea_n   = (const float*)d_in[12];
    const float* W_an1_s  = (const float*)d_in[13];
    const float* W_an1_n  = (const float*)d_in[14];
    const float* W_an2_s  = (const float*)d_in[15];
    const float* W_an2_n  = (const float*)d_in[16];
    const float* Wq       = (const float*)d_in[17];
    const float* Wk       = (const float*)d_in[18];
    const float* Wv       = (const float*)d_in[19];
    const float* W_out    = (const float*)d_in[20];

    const int* H0 = Hm;          const int* H1 = Hm + EE;
    const int* src = rei;        const int* dst = rei + EE;
    const int* src_lg = lgei;    const int* dst_lg = lgei + LGE;

    // -------- workspace carve-up (floats) --------
    float* ws = (float*)d_ws;
    size_t o = 0;
    auto take = [&](size_t n) { float* p = ws + o; o += n; return p; };
    float* B1   = take((size_t)EE * HID);     // h1
    float* B2   = take((size_t)EE * HID);     // nbr(et) then h2
    float* B3   = take((size_t)EE * HID);     // nbr(h1)
    float* degE = take(EE);                   // line-graph deg, later raw node deg
    float* degN = take(NN);                   // H-incidence deg
    float* AGG  = take((size_t)NN * HID);     // EdgeToNode agg
    float* LEAK = take((size_t)NN * HID);     // leaky(node_e)
    float* EREP = take((size_t)NN * HID);     // edge_repr
    float* NBRD = take((size_t)NN * HID);     // nbr(edge_repr)
    float* AEDG = take((size_t)NN * HID);     // aggr_edge
    float* NBX  = take((size_t)NN * FND);     // nbr(x)
    float* HN1  = take((size_t)NN * HID);
    float* NBG  = take((size_t)NN * HID);     // nbr(hn1)
    float* HN   = take((size_t)NN * HID);
    float* Qb   = take((size_t)NN * HID);
    float* Kb   = take((size_t)NN * HID);
    float* Vb   = take((size_t)NN * HID);
    float* OUTB = take((size_t)NN * HID);     // hn + attn
    float* SC   = take(EE);
    float* MB   = take(NN);
    float* EX   = take(EE);
    float* DEN  = take(NN);
    float* LGT  = take((size_t)NN * OUTD);

    auto nb = [](long n) { return (unsigned)((n + 255) / 256); };
    // fused C = act(A1@W1 (+ A2@W2)); Nc in {64,128}
    auto gemm2 = [&](const float* A1, const float* W1,
                     const float* A2, const float* W2, float* C,
                     int M, int K, int Nc, int act) {
        int blocks = ((M >> 4) + 7) / 8;
        if (Nc == 128 && A2)
            k_gemm2_f32_wmma<8, true><<<blocks, 256, 0, stream>>>(A1, W1, A2, W2, C, M, K, act);
        else if (Nc == 128)
            k_gemm2_f32_wmma<8, false><<<blocks, 256, 0, stream>>>(A1, W1, A1, W1, C, M, K, act);
        else
            k_gemm2_f32_wmma<4, false><<<blocks, 256, 0, stream>>>(A1, W1, A1, W1, C, M, K, act);
    };

    // ======== Phase A: TSA layer 1 on line graph (T -> HID, relu) ========
    k_fill<<<nb(EE), 256, 0, stream>>>(degE, 0.f, EE);
    k_fill<<<nb((long)EE * TT), 256, 0, stream>>>(B2, 0.f, (long)EE * TT);
    k_deg<<<nb(LGE), 256, 0, stream>>>(dst_lg, degE, LGE);
    k_scatter_add<<<nb((long)LGE * (TT / 4)), 256, 0, stream>>>(et, src_lg, dst_lg, B2, LGE, TT, 4);
    k_div_deg<<<nb((long)EE * TT), 256, 0, stream>>>(B2, degE, EE, 6);
    gemm2(et, W_tsa1_s, B2, W_tsa1_n, B1, EE, TT, HID, 1);          // h1

    // ======== Phase B: TSA layer 2 (HID -> HID) ========
    k_fill<<<nb((long)EE * HID), 256, 0, stream>>>(B3, 0.f, (long)EE * HID);
    k_scatter_add<<<nb((long)LGE * (HID / 4)), 256, 0, stream>>>(B1, src_lg, dst_lg, B3, LGE, HID, 5);
    k_div_deg<<<nb((long)EE * HID), 256, 0, stream>>>(B3, degE, EE, 7);
    gemm2(B1, W_tsa2_s, B3, W_tsa2_n, B2, EE, HID, HID, 0);         // h2 = B2

    // ======== Phase C: EdgeToNodeConv + EdgeGCN output ========
    k_fill<<<nb(NN), 256, 0, stream>>>(degN, 0.f, NN);
    k_fill<<<nb((long)NN * HID), 256, 0, stream>>>(AGG, 0.f, (long)NN * HID);
    k_deg<<<nb(EE), 256, 0, stream>>>(H0, degN, EE);
    k_deg<<<nb(EE), 256, 0, stream>>>(H1, degN, EE);
    k_scatter_add<<<nb((long)EE * (HID / 4)), 256, 0, stream>>>(B2, nullptr, H0, AGG, EE, HID, 5);
    k_scatter_add<<<nb((long)EE * (HID / 4)), 256, 0, stream>>>(B2, nullptr, H1, AGG, EE, HID, 5);
    k_div_deg<<<nb((long)NN * HID), 256, 0, stream>>>(AGG, degN, NN, 7);
    gemm2(AGG, W_etn, nullptr, nullptr, LEAK, NN, HID, HID, 2);     // leaky(node_e)
    gemm2(LEAK, W_eg_lin, nullptr, nullptr, EREP, NN, HID, HID, 0); // edge_repr

    // ======== Phase D: edge_aggr SAGE over raw graph ========
    k_fill<<<nb(NN), 256, 0, stream>>>(degE, 0.f, NN);              // reuse: raw node deg
    k_deg<<<nb(EE), 256, 0, stream>>>(dst, degE, EE);
    k_fill<<<nb((long)NN * HID), 256, 0, stream>>>(NBRD, 0.f, (long)NN * HID);
    k_scatter_add<<<nb((long)EE * (HID / 4)), 256, 0, stream>>>(EREP, src, dst, NBRD, EE, HID, 5);
    k_div_deg<<<nb((long)NN * HID), 256, 0, stream>>>(NBRD, degE, NN, 7);
    gemm2(EREP, W_ea_s, NBRD, W_ea_n, AEDG, NN, HID, HID, 0);       // aggr_edge

    // ======== Phase E: attr_node_model (2-layer SAGE on x) ========
    k_fill<<<nb((long)NN * FND), 256, 0, stream>>>(NBX, 0.f, (long)NN * FND);
    k_scatter_add<<<nb((long)EE * (FND / 4)), 256, 0, stream>>>(x, src, dst, NBX, EE, FND, 6);
    k_div_deg<<<nb((long)NN * FND), 256, 0, stream>>>(NBX, degE, NN, 8);
    gemm2(x, W_an1_s, NBX, W_an1_n, HN1, NN, FND, HID, 1);          // relu
    k_fill<<<nb((long)NN * HID), 256, 0, stream>>>(NBG, 0.f, (long)NN * HID);
    k_scatter_add<<<nb((long)EE * (HID / 4)), 256, 0, stream>>>(HN1, src, dst, NBG, EE, HID, 5);
    k_div_deg<<<nb((long)NN * HID), 256, 0, stream>>>(NBG, degE, NN, 7);
    gemm2(HN1, W_an2_s, NBG, W_an2_n, HN, NN, HID, HID, 0);         // hn

    // ======== Phase F: MixAttention ========
    gemm2(HN,   Wq, nullptr, nullptr, Qb, NN, HID, HID, 0);
    gemm2(AEDG, Wk, nullptr, nullptr, Kb, NN, HID, HID, 0);
    gemm2(AEDG, Wv, nullptr, nullptr, Vb, NN, HID, HID, 0);
    k_edge_score<<<nb((long)EE * 32), 256, 0, stream>>>(Qb, Kb, src, dst, SC, EE);
    k_fill<<<nb(NN), 256, 0, stream>>>(MB, -INFINITY, NN);
    k_fill<<<nb(NN), 256, 0, stream>>>(DEN, 0.f, NN);
    k_seg_max<<<nb(EE), 256, 0, stream>>>(SC, dst, MB, EE);
    k_fix_m<<<nb(NN), 256, 0, stream>>>(MB, NN);
    k_exp_den<<<nb(EE), 256, 0, stream>>>(SC, dst, MB, EX, DEN, EE);
    k_copy<<<nb((long)NN * HID), 256, 0, stream>>>(OUTB, HN, (long)NN * HID);
    k_attn_scatter<<<nb((long)EE * (HID / 4)), 256, 0, stream>>>(EX, DEN, Vb, src, dst, OUTB, EE);

    // ======== Phase G: classifier + log_softmax ========
    gemm2(OUTB, W_out, nullptr, nullptr, LGT, NN, HID, OUTD, 0);
    k_logsoftmax<<<nb((long)NN * 32), 256, 0, stream>>>(LGT, (float*)d_out, NN);
}